// NetVLAD_cent_21861383536707
// MI455X (gfx1250) — compile-verified
//
#include <hip/hip_runtime.h>
#include <hip/hip_bf16.h>
#include <cstddef>

typedef __attribute__((ext_vector_type(2))) float v2f;
typedef __attribute__((ext_vector_type(8))) float v8f;
typedef __attribute__((ext_vector_type(4))) int   v4i;

#if defined(__has_builtin)
#  if __has_builtin(__builtin_amdgcn_global_load_async_to_lds_b128)
#    define HAVE_ASYNC_LDS 1
#  endif
#endif
#ifndef HAVE_ASYNC_LDS
#  define HAVE_ASYNC_LDS 0
#endif

#define NN     64
#define CC     128
#define KK     64
#define PP     4096
#define PT     32        // p-tile width
#define SPLITS 8
#define PCHUNK (PP / SPLITS)   // 512 -> 16 tiles per block
#define NTILES (PCHUNK / PT)   // 16
#define ALPHA  50.0f
#define EPSN   1e-12f
#define VLAD_ELEMS (NN * KK * CC)        // 524288
#define LOGOFF ((size_t)NN * KK * CC)    // vlad first, logits second in d_out

// LDS float offsets (double-buffered x tile)
#define XS_LD  36
#define SA_LD  37
#define OFF_XS0  0
#define OFF_XS1  (OFF_XS0 + CC * XS_LD)       // 4608
#define OFF_SA   (OFF_XS1 + CC * XS_LD)       // 9216
#define OFF_WN   (OFF_SA + KK * SA_LD)        // 11584
#define OFF_INV  (OFF_WN + CC * KK)           // 19776
#define OFF_CSUM (OFF_INV + PT)               // 19808
#define OFF_SSUM (OFF_CSUM + PT)              // 19840
#define OFF_BIAS (OFF_SSUM + KK)              // 19904
#define SMEM_FLOATS (OFF_BIAS + KK)           // 19968 floats = 79872 B

__global__ void zero_ws_kernel(float* __restrict__ p, int nel) {
    int i = blockIdx.x * 256 + threadIdx.x;
    if (i < nel) p[i] = 0.0f;
}

#if HAVE_ASYNC_LDS
typedef __attribute__((address_space(1))) v4i* as1_v4i;
typedef __attribute__((address_space(3))) v4i* as3_v4i;
// Issue this thread's share (4 x b128 = 64B) of one 128x32 tile load.
__device__ __forceinline__ void issue_tile_async(const float* __restrict__ xn_ptr,
                                                 int pbase, float* __restrict__ dst,
                                                 int tid) {
    #pragma unroll
    for (int i = 0; i < 4; ++i) {
        int idx4 = tid + i * 256;          // 1024 float4 slots
        int c  = idx4 >> 3;
        int pq = idx4 & 7;
        as1_v4i g = (as1_v4i)(xn_ptr + ((size_t)c * PP + pbase + pq * 4));
        as3_v4i l = (as3_v4i)(&dst[c * XS_LD + pq * 4]);
        __builtin_amdgcn_global_load_async_to_lds_b128(g, l, 0, 0);
    }
}
#endif

__device__ __forceinline__ void load_tile_sync(const float* __restrict__ xn_ptr,
                                               int pbase, float* __restrict__ dst,
                                               int tid) {
    #pragma unroll
    for (int i = 0; i < 4; ++i) {
        int idx4 = tid + i * 256;
        int c  = idx4 >> 3;
        int pq = idx4 & 7;
        const float4 v = *reinterpret_cast<const float4*>(
            xn_ptr + ((size_t)c * PP + pbase + pq * 4));
        *reinterpret_cast<float4*>(&dst[c * XS_LD + pq * 4]) = v;
    }
}

__global__ __launch_bounds__(256) void netvlad_main_kernel(
    const float* __restrict__ x,     // [N, C, P]
    const float* __restrict__ Wm,    // [C, K]
    const float* __restrict__ bias,  // [K]
    float* __restrict__ out,         // vlad region + logits region
    float* __restrict__ vlad_ws,     // [N, K, C] partial accum
    float* __restrict__ ssum_ws)     // [N, K]
{
    __shared__ __align__(16) float smem[SMEM_FLOATS];
    float* xs0    = smem + OFF_XS0;    // [128][36] x tile buffer 0
    float* xs1    = smem + OFF_XS1;    // [128][36] x tile buffer 1
    float* sa     = smem + OFF_SA;     // [64][37]  raw logits -> exp -> sa'
    float* wn     = smem + OFF_WN;     // [128][64] column-normalized W
    float* invn_s = smem + OFF_INV;    // [32]
    float* csum_s = smem + OFF_CSUM;   // [32]
    float* ssum_s = smem + OFF_SSUM;   // [64]
    float* bias_s = smem + OFF_BIAS;   // [64]

    const int tid  = threadIdx.x;
    const int lane = tid & 31;
    const int w    = tid >> 5;          // wave id 0..7 (wave32)
    const int lm   = lane & 15;         // M/N index within 16x16 tile
    const int bk   = (lane >> 4) << 1;  // reduction sub-index base: 0 or 2
    const int khalf = (lane >> 4) << 3; // +8 row offset for high half lanes

    const int blk   = blockIdx.x;
    const int n     = blk / SPLITS;
    const int split = blk % SPLITS;
    const int pbase0 = split * PCHUNK;

    const int k0  = (w >> 1) * 16;
    const int p0  = (w & 1) * 16;
    const int cb0 = (w & 1) * 64;

    const float* xn_ptr = x + (size_t)n * CC * PP;

#if HAVE_ASYNC_LDS
    // Prefetch tile 0 asynchronously; overlaps with the Wn preamble below.
    issue_tile_async(xn_ptr, pbase0, xs0, tid);
#endif

    // ---- load W into LDS, column-normalize (over C) ----
    for (int idx = tid; idx < CC * KK; idx += 256) wn[idx] = Wm[idx];
    if (tid < KK) { ssum_s[tid] = 0.0f; bias_s[tid] = bias[tid]; }
    __syncthreads();
    if (tid < KK) {
        float s = 0.0f;
        for (int c = 0; c < CC; ++c) { float v = wn[c * KK + tid]; s += v * v; }
        float inv = 1.0f / fmaxf(sqrtf(s), EPSN);
        for (int c = 0; c < CC; ++c) wn[c * KK + tid] *= inv;
    }

    // vlad accumulators: 4 tiles of 16x16 per wave (k band x 4 c tiles)
    v8f acc0 = {0,0,0,0,0,0,0,0};
    v8f acc1 = {0,0,0,0,0,0,0,0};
    v8f acc2 = {0,0,0,0,0,0,0,0};
    v8f acc3 = {0,0,0,0,0,0,0,0};

    for (int pt = 0; pt < NTILES; ++pt) {
        const int pbase = pbase0 + pt * PT;
        float* xs = (pt & 1) ? xs1 : xs0;

        __syncthreads();   // previous tile's GEMM2 reads done; other buffer free

#if HAVE_ASYNC_LDS
        if (pt + 1 < NTILES) {
            // Kick next tile into the other buffer, then wait for current tile:
            // the 4 loads just issued may remain outstanding; everything older
            // (current tile's group) completes in order.
            issue_tile_async(xn_ptr, pbase + PT, (pt & 1) ? xs0 : xs1, tid);
            asm volatile("s_wait_asynccnt 0x4" ::: "memory");
        } else {
            asm volatile("s_wait_asynccnt 0x0" ::: "memory");
        }
#else
        load_tile_sync(xn_ptr, pbase, xs, tid);
#endif
        __syncthreads();

        // ---- per-column inverse L2 norm over channels ----
        if (tid < PT) {
            float s = 0.0f;
            for (int c = 0; c < CC; ++c) { float v = xs[c * XS_LD + tid]; s += v * v; }
            invn_s[tid] = 1.0f / fmaxf(sqrtf(s), EPSN);
        }
        __syncthreads();

        // ---- GEMM1: raw[k,p] = sum_c Wn[c,k] * x[c,p]  (WMMA f32 16x16x4) ----
        v8f rr = {0,0,0,0,0,0,0,0};
        #pragma unroll
        for (int c0 = 0; c0 < CC; c0 += 4) {
            v2f a, b;
            a.x = wn[(c0 + bk    ) * KK + k0 + lm];
            a.y = wn[(c0 + bk + 1) * KK + k0 + lm];
            b.x = xs[(c0 + bk    ) * XS_LD + p0 + lm];
            b.y = xs[(c0 + bk + 1) * XS_LD + p0 + lm];
            rr = __builtin_amdgcn_wmma_f32_16x16x4_f32(
                false, a, false, b, (short)0, rr, false, false);
        }

        // ---- store raw to LDS; scaled logits to global (reference output 2) ----
        {
            float iv = invn_s[p0 + lm];
            #pragma unroll
            for (int r8 = 0; r8 < 8; ++r8) {
                int k = k0 + khalf + r8;
                float raw = rr[r8];
                sa[k * SA_LD + p0 + lm] = raw;
                out[LOGOFF + ((size_t)(n * KK + k)) * PP + pbase + p0 + lm] = raw * iv;
            }
        }
        __syncthreads();

        // ---- softmax over k per column p (threads 0..31 own one column) ----
        if (tid < PT) {
            const int p = tid;
            const float iv = invn_s[p];
            float mx = -3.0e38f;
            for (int k = 0; k < KK; ++k) {
                float z = (sa[k * SA_LD + p] * iv + bias_s[k]) * ALPHA;
                mx = fmaxf(mx, z);
            }
            float s = 0.0f;
            for (int k = 0; k < KK; ++k) {
                float z = (sa[k * SA_LD + p] * iv + bias_s[k]) * ALPHA;
                float e = __expf(z - mx);
                sa[k * SA_LD + p] = e;
                s += e;
            }
            csum_s[p] = s;
        }
        __syncthreads();

        // ---- sa' = (e/colsum)*invn ; accumulate ssum[k] (threads 0..63) ----
        if (tid < KK) {
            float rowacc = 0.0f;
            for (int p = 0; p < PT; ++p) {
                float v = sa[tid * SA_LD + p] / csum_s[p];
                rowacc += v;
                sa[tid * SA_LD + p] = v * invn_s[p];
            }
            ssum_s[tid] += rowacc;
        }
        __syncthreads();

        // ---- GEMM2: vlad[k,c] += sum_p sa'[k,p] * x[c,p] ----
        #pragma unroll
        for (int s8 = 0; s8 < 8; ++s8) {
            const int pp = s8 * 4;
            v2f a;
            a.x = sa[(k0 + lm) * SA_LD + pp + bk    ];
            a.y = sa[(k0 + lm) * SA_LD + pp + bk + 1];
            v2f b;
            b.x = xs[(cb0 +  0 + lm) * XS_LD + pp + bk    ];
            b.y = xs[(cb0 +  0 + lm) * XS_LD + pp + bk + 1];
            acc0 = __builtin_amdgcn_wmma_f32_16x16x4_f32(false, a, false, b, (short)0, acc0, false, false);
            b.x = xs[(cb0 + 16 + lm) * XS_LD + pp + bk    ];
            b.y = xs[(cb0 + 16 + lm) * XS_LD + pp + bk + 1];
            acc1 = __builtin_amdgcn_wmma_f32_16x16x4_f32(false, a, false, b, (short)0, acc1, false, false);
            b.x = xs[(cb0 + 32 + lm) * XS_LD + pp + bk    ];
            b.y = xs[(cb0 + 32 + lm) * XS_LD + pp + bk + 1];
            acc2 = __builtin_amdgcn_wmma_f32_16x16x4_f32(false, a, false, b, (short)0, acc2, false, false);
            b.x = xs[(cb0 + 48 + lm) * XS_LD + pp + bk    ];
            b.y = xs[(cb0 + 48 + lm) * XS_LD + pp + bk + 1];
            acc3 = __builtin_amdgcn_wmma_f32_16x16x4_f32(false, a, false, b, (short)0, acc3, false, false);
        }
    }

    // ---- flush vlad partials (atomic f32 add) and ssum partials ----
    {
        float* vbase = vlad_ws + (size_t)n * KK * CC;
        #pragma unroll
        for (int r8 = 0; r8 < 8; ++r8) {
            int k = k0 + khalf + r8;
            atomicAdd(&vbase[k * CC + cb0 +  0 + lm], acc0[r8]);
            atomicAdd(&vbase[k * CC + cb0 + 16 + lm], acc1[r8]);
            atomicAdd(&vbase[k * CC + cb0 + 32 + lm], acc2[r8]);
            atomicAdd(&vbase[k * CC + cb0 + 48 + lm], acc3[r8]);
        }
    }
    __syncthreads();
    if (tid < KK) atomicAdd(&ssum_ws[n * KK + tid], ssum_s[tid]);
}

__global__ __launch_bounds__(256) void netvlad_finish_kernel(
    const float* __restrict__ Wm,       // [C, K]
    const float* __restrict__ vlad_ws,  // [N, K, C]
    const float* __restrict__ ssum_ws,  // [N, K]
    float* __restrict__ out)            // vlad region at offset 0
{
    __shared__ __align__(16) float vs[KK * 129];
    __shared__ float ss[KK];
    __shared__ float rs[KK];
    __shared__ float gsum;

    const int n = blockIdx.x;
    const int tid = threadIdx.x;

    if (tid < KK) ss[tid] = ssum_ws[n * KK + tid];
    if (tid == 0) gsum = 0.0f;
    __syncthreads();

    // vlad[k,c] = accum - ssum[k] * W[c,k]
    for (int idx = tid; idx < KK * CC; idx += 256) {
        int k = idx >> 7, c = idx & 127;
        float v = vlad_ws[(size_t)n * KK * CC + idx] - ss[k] * Wm[c * KK + k];
        vs[k * 129 + c] = v;
    }
    __syncthreads();

    // intra-normalize each cluster row over C
    if (tid < KK) {
        float s = 0.0f;
        for (int c = 0; c < CC; ++c) { float v = vs[tid * 129 + c]; s += v * v; }
        rs[tid] = 1.0f / fmaxf(sqrtf(s), EPSN);
    }
    __syncthreads();

    float part = 0.0f;
    for (int idx = tid; idx < KK * CC; idx += 256) {
        int k = idx >> 7, c = idx & 127;
        float v = vs[k * 129 + c] * rs[k];
        vs[k * 129 + c] = v;
        part += v * v;
    }
    atomicAdd(&gsum, part);
    __syncthreads();

    float gsc = 1.0f / fmaxf(sqrtf(gsum), EPSN);
    for (int idx = tid; idx < KK * CC; idx += 256) {
        int k = idx >> 7, c = idx & 127;
        out[(size_t)n * KK * CC + idx] = vs[k * 129 + c] * gsc;
    }
}

extern "C" void kernel_launch(void* const* d_in, const int* in_sizes, int n_in,
                              void* d_out, int out_size, void* d_ws, size_t ws_size,
                              hipStream_t stream) {
    const float* x    = (const float*)d_in[0];   // [64,128,64,64]
    const float* Wm   = (const float*)d_in[1];   // [128,64]
    const float* bias = (const float*)d_in[2];   // [64]
    float* out = (float*)d_out;

    float* vlad_ws = (float*)d_ws;               // N*K*C floats
    float* ssum_ws = vlad_ws + VLAD_ELEMS;       // N*K floats
    const int nz = VLAD_ELEMS + NN * KK;

    zero_ws_kernel<<<(nz + 255) / 256, 256, 0, stream>>>(vlad_ws, nz);
    netvlad_main_kernel<<<NN * SPLITS, 256, 0, stream>>>(x, Wm, bias, out, vlad_ws, ssum_ws);
    netvlad_finish_kernel<<<NN, 256, 0, stream>>>(Wm, vlad_ws, ssum_ws, out);
}